// IMU_NN_67422396612813
// MI455X (gfx1250) — compile-verified
//
#include <hip/hip_runtime.h>
#include <math.h>

// ---- problem constants ----
static constexpr int cL = 32, cB = 4, cD = 64, cHEADS = 4, cDK = 16;
static constexpr float EPS = 1e-5f;
static constexpr float BN_INV = 0.99999500003749969f; // 1/sqrt(1+1e-5)

// ---- workspace layout (float offsets) ----
static constexpr size_t OFF_WEFF = 0;              // DEPTH*64*128 = 16384
static constexpr size_t OFF_BEFF = 16384;          // DEPTH*128    = 256
static constexpr size_t OFF_X    = 32768;          // 2097152  (X for LN1, X2 for LN2)
static constexpr size_t OFF_K    = OFF_X  + 2097152;
static constexpr size_t OFF_V    = OFF_K  + 2097152;
static constexpr size_t OFF_CK   = OFF_V  + 2097152;
static constexpr size_t OFF_AP   = OFF_CK + 2097152;  // 65536
static constexpr size_t OFF_AT   = OFF_AP + 65536;    // 65536

typedef __attribute__((ext_vector_type(2))) float v2f;
typedef __attribute__((ext_vector_type(8))) float v8f;

__device__ __forceinline__ v8f wmma_f32_4(v2f a, v2f b, v8f c) {
  // V_WMMA_F32_16X16X4_F32 : D = A(16x4,f32) * B(4x16,f32) + C(16x16,f32)
  return __builtin_amdgcn_wmma_f32_16x16x4_f32(false, a, false, b, (short)0, c,
                                               false, false);
}

__device__ __forceinline__ float lrelu(float v) { return v > 0.f ? v : 0.01f * v; }

// ===================================================================
// Precompute fused KV weights: Weff[c][m] (64x128 per depth), beff[m]
// m = n*32 + s*16 + dk ; s=0 -> K path, s=1 -> V path
// Weff = conv1_w(k/v slice) composed with wk/wv ; beff from conv1_b.
// ===================================================================
__global__ void k_weff(const float* __restrict__ conv1_w,
                       const float* __restrict__ conv1_b,
                       const float* __restrict__ wk,
                       const float* __restrict__ wv,
                       float* __restrict__ ws) {
  int bid = blockIdx.x;          // d*128 + m
  int d = bid >> 7, m = bid & 127;
  int n = m >> 5, s = (m >> 4) & 1, dk = m & 15;
  int c = threadIdx.x;           // 0..63
  int orow = (s == 0) ? 64 : 128;          // offset into 3D qkv rows (skip q)
  const float* c1w = conv1_w + ((size_t)(d * 4 + n) * 192 + orow) * 64;
  const float* proj = ((s == 0) ? wk : wv) + (size_t)(d * 4 + n) * 64 * 16;
  float acc = 0.f;
  for (int cp = 0; cp < 64; ++cp)
    acc += c1w[cp * 64 + c] * proj[cp * 16 + dk];
  ws[OFF_WEFF + (size_t)d * 8192 + (size_t)c * 128 + m] = acc;
  if (c == 0) {
    const float* c1b = conv1_b + (size_t)(d * 4 + n) * 192 + orow;
    float bacc = 0.f;
    for (int cp = 0; cp < 64; ++cp) bacc += c1b[cp] * proj[cp * 16 + dk];
    ws[OFF_BEFF + (size_t)d * 128 + m] = bacc;
  }
}

// ===================================================================
// Feature generator: 4x (conv3x3 + bias)*BN_INV -> lrelu, then + pos.
// One block per frame (l,b); thread = pixel. Writes all of d_out.
// ===================================================================
__device__ __forceinline__ float conv3x3(const float* sbuf, const float* wrow,
                                         int cin, int h, int w) {
  float acc = 0.f;
  for (int ci = 0; ci < cin; ++ci)
    for (int dy = 0; dy < 3; ++dy) {
      int yy = h + dy - 1; if ((unsigned)yy >= 16u) continue;
      for (int dx = 0; dx < 3; ++dx) {
        int xx = w + dx - 1; if ((unsigned)xx >= 16u) continue;
        acc += sbuf[ci * 256 + yy * 16 + xx] * wrow[ci * 9 + dy * 3 + dx];
      }
    }
  return acc;
}

__global__ void k_featgen(const float* __restrict__ x,
                          const float* __restrict__ gw1, const float* __restrict__ gb1,
                          const float* __restrict__ gw2, const float* __restrict__ gb2,
                          const float* __restrict__ gw3, const float* __restrict__ gb3,
                          const float* __restrict__ gw4, const float* __restrict__ gb4,
                          float* __restrict__ out) {
  __shared__ float s_in[3 * 256];
  __shared__ float s1[4 * 256];
  __shared__ float s2[6 * 256];
  __shared__ float s3[8 * 256];
  __shared__ float w1s[108], b1s[4], w2s[216], b2s[6], w3s[432], b3s[8];
  __shared__ float w4s[4608], b4s[64];
  int f = blockIdx.x;            // l*4 + b
  int l = f >> 2, b = f & 3;
  int t = threadIdx.x;           // pixel
  for (int i = t; i < 108; i += 256) w1s[i] = gw1[i];
  for (int i = t; i < 4;   i += 256) b1s[i] = gb1[i];
  for (int i = t; i < 216; i += 256) w2s[i] = gw2[i];
  for (int i = t; i < 6;   i += 256) b2s[i] = gb2[i];
  for (int i = t; i < 432; i += 256) w3s[i] = gw3[i];
  for (int i = t; i < 8;   i += 256) b3s[i] = gb3[i];
  for (int i = t; i < 4608; i += 256) w4s[i] = gw4[i];
  for (int i = t; i < 64;  i += 256) b4s[i] = gb4[i];
  for (int i = t; i < 3 * 256; i += 256) s_in[i] = x[(size_t)f * 768 + i];
  __syncthreads();
  int h = t >> 4, w = t & 15;
  for (int co = 0; co < 4; ++co) {
    float a = (conv3x3(s_in, w1s + co * 27, 3, h, w) + b1s[co]) * BN_INV;
    s1[co * 256 + t] = lrelu(a);
  }
  __syncthreads();
  for (int co = 0; co < 6; ++co) {
    float a = (conv3x3(s1, w2s + co * 36, 4, h, w) + b2s[co]) * BN_INV;
    s2[co * 256 + t] = lrelu(a);
  }
  __syncthreads();
  for (int co = 0; co < 8; ++co) {
    float a = (conv3x3(s2, w3s + co * 54, 6, h, w) + b3s[co]) * BN_INV;
    s3[co * 256 + t] = lrelu(a);
  }
  __syncthreads();
  for (int co = 0; co < 64; ++co) {
    float a = (conv3x3(s3, w4s + co * 72, 8, h, w) + b4s[co]) * BN_INV;
    a = lrelu(a);
    // positional table: angle = l / 10000^(2*(co/2)/64); even l -> sin, odd l -> cos
    float e = (float)(2 * (co >> 1)) * (1.0f / 64.0f);
    float ang = (float)l * expf(-e * 9.210340371976184f); // ln(10000)
    float pos = ((l & 1) == 0) ? sinf(ang) : cosf(ang);
    out[(size_t)l * 65536 + (size_t)b * 16384 + (size_t)co * 256 + t] = a + pos;
  }
}

// ===================================================================
// LayerNorm over (D,H,W) per (l,b).  mode 0: write X[p1*64+c], p1=(l,b,hw)
//                                    mode 1: write X2[(b*16384+c*256+hw)*32+l]
// ===================================================================
__global__ void k_layernorm(const float* __restrict__ xin,
                            const float* __restrict__ lnw,
                            const float* __restrict__ lnb,
                            float* __restrict__ dst, int mode) {
  __shared__ float red[256];
  __shared__ float sstat[2];
  int f = blockIdx.x;            // l*4 + b
  int l = f >> 2, b = f & 3;
  const float* xp = xin + (size_t)f * 16384;
  int t = threadIdx.x;
  float s = 0.f;
  for (int e = 0; e < 64; ++e) s += xp[e * 256 + t];
  red[t] = s; __syncthreads();
  for (int off = 128; off > 0; off >>= 1) {
    if (t < off) red[t] += red[t + off];
    __syncthreads();
  }
  if (t == 0) sstat[0] = red[0] * (1.0f / 16384.0f);
  __syncthreads();
  float mu = sstat[0];
  float s2 = 0.f;
  for (int e = 0; e < 64; ++e) { float d = xp[e * 256 + t] - mu; s2 += d * d; }
  red[t] = s2; __syncthreads();
  for (int off = 128; off > 0; off >>= 1) {
    if (t < off) red[t] += red[t + off];
    __syncthreads();
  }
  if (t == 0) sstat[1] = red[0] * (1.0f / 16384.0f);
  __syncthreads();
  float inv = rsqrtf(sstat[1] + EPS);
  for (int e = 0; e < 64; ++e) {
    int idx = e * 256 + t;       // c*256 + hw
    float val = (xp[idx] - mu) * inv * lnw[idx] + lnb[idx];
    if (mode == 0)
      dst[((size_t)f * 256 + t) * 64 + e] = val;
    else
      dst[((size_t)b * 16384 + idx) * 32 + l] = val;
  }
}

// ===================================================================
// KV GEMM via WMMA f32 16x16x4 : X(32768x64) @ Weff(64x128) + beff
// col m = n*32 + s*16 + dk; s=0 -> k_buf[(b,l,n*16+dk,hw)]
//                           s=1 -> v_buf[(b,n,dk,hw,l)]
// ===================================================================
__global__ void k_kvgemm(const float* __restrict__ X,
                         const float* __restrict__ weff,
                         const float* __restrict__ beff,
                         float* __restrict__ kbuf,
                         float* __restrict__ vbuf) {
  __shared__ float sA[16 * 64];
  int mt = blockIdx.x;           // 0..2047
  int t = threadIdx.x;           // 0..127
  for (int j = 0; j < 8; ++j) sA[t + j * 128] = X[(size_t)mt * 1024 + t + j * 128];
  __syncthreads();
  int wave = t >> 5, lane = t & 31;
  int lr = lane & 15;
  int kh = (lane >> 4) << 1;
  int rb = (lane >> 4) << 3;
  for (int nt = wave * 2; nt < wave * 2 + 2; ++nt) {
    int col = nt * 16 + lr;
    v8f acc = {};
    for (int k0 = 0; k0 < 64; k0 += 4) {
      v2f a, bf;
      a.x = sA[lr * 64 + k0 + kh];
      a.y = sA[lr * 64 + k0 + kh + 1];
      bf.x = weff[(size_t)(k0 + kh) * 128 + col];
      bf.y = weff[(size_t)(k0 + kh + 1) * 128 + col];
      acc = wmma_f32_4(a, bf, acc);
    }
    float bias = beff[col];
    int n = col >> 5, ssel = (col >> 4) & 1, dk = col & 15;
    for (int g = 0; g < 8; ++g) {
      int p1 = mt * 16 + g + rb;                 // (l,b,hw) row
      int l = p1 >> 10, b = (p1 >> 8) & 3, hw = p1 & 255;
      float val = acc[g] + bias;
      if (ssel == 0)
        kbuf[((size_t)(b * 32 + l) * 64 + n * 16 + dk) * 256 + hw] = val;
      else
        vbuf[(((size_t)(b * 4 + n) * 16 + dk) * 256 + hw) * 32 + l] = val;
    }
  }
}

// ===================================================================
// Per-head 5x5 conv on K frames (q path cancels in softmax -> skipped).
// Block per (b,l,n); thread = pixel; writes ck[(b,n,dk,hw,l)].
// ===================================================================
__global__ void k_headconv(const float* __restrict__ kbuf,
                           const float* __restrict__ conv2_w,
                           float* __restrict__ ck) {
  __shared__ float sIn[16 * 256];
  __shared__ float sW[16 * 16 * 25];
  int bid = blockIdx.x;          // ((b*32+l)*4 + n)
  int n = bid & 3, l = (bid >> 2) & 31, b = bid >> 7;
  int t = threadIdx.x;
  const float* src = kbuf + ((size_t)(b * 32 + l) * 64 + n * 16) * 256;
  for (int i = t; i < 4096; i += 256) sIn[i] = src[i];
  const float* wsrc = conv2_w + (size_t)n * 6400;
  for (int i = t; i < 6400; i += 256) sW[i] = wsrc[i];
  __syncthreads();
  int h = t >> 4, w = t & 15;
  for (int dko = 0; dko < 16; ++dko) {
    float acc = 0.f;
    for (int dki = 0; dki < 16; ++dki) {
      const float* wp = &sW[(dko * 16 + dki) * 25];
      const float* ip = &sIn[dki * 256];
      for (int dy = 0; dy < 5; ++dy) {
        int yy = h + dy - 2; if ((unsigned)yy >= 16u) continue;
        for (int dx = 0; dx < 5; ++dx) {
          int xx = w + dx - 2; if ((unsigned)xx >= 16u) continue;
          acc += ip[yy * 16 + xx] * wp[dy * 5 + dx];
        }
      }
    }
    ck[(((size_t)(b * 4 + n) * 16 + dko) * 256 + t) * 32 + l] = acc;
  }
}

// ===================================================================
// softmax over L (=32 = wave32 lanes) of ck row, dot with v row.
// softmax(cq_i + ck_j + bias) == softmax(ck_j) exactly -> i/bias-free.
// One wave per row; __shfl_xor reductions (wave32).
// ===================================================================
__global__ void k_softmax_av(const float* __restrict__ ck,
                             const float* __restrict__ vbuf,
                             float* __restrict__ ap) {
  int row = blockIdx.x * 8 + (threadIdx.x >> 5);   // (b,c,hw) row, 65536 total
  int lane = threadIdx.x & 31;
  float cv = ck[(size_t)row * 32 + lane];
  float m = cv;
  for (int off = 16; off > 0; off >>= 1) m = fmaxf(m, __shfl_xor(m, off, 32));
  float e = expf(cv - m);
  float s = e;
  for (int off = 16; off > 0; off >>= 1) s += __shfl_xor(s, off, 32);
  float wv = (e / s) * vbuf[(size_t)row * 32 + lane];
  for (int off = 16; off > 0; off >>= 1) wv += __shfl_xor(wv, off, 32);
  if (lane == 0) ap[row] = wv;
}

// ===================================================================
// W0 projection via WMMA: attn_pre(1024x64) @ w0(64x64) -> attn(b,o,hw)
// ===================================================================
__global__ void k_w0gemm(const float* __restrict__ ap,
                         const float* __restrict__ w0,
                         float* __restrict__ attn) {
  __shared__ float sA[16 * 64];
  int mt = blockIdx.x;           // 0..63
  int t = threadIdx.x;           // 0..127
  for (int j = 0; j < 8; ++j) {
    int e = t + j * 128;
    int r = e >> 6, c = e & 63;
    int p = mt * 16 + r;
    int b = p >> 8, hw = p & 255;
    sA[e] = ap[(size_t)b * 16384 + (size_t)c * 256 + hw];
  }
  __syncthreads();
  int wave = t >> 5, lane = t & 31;
  int lr = lane & 15, kh = (lane >> 4) << 1, rb = (lane >> 4) << 3;
  int col = wave * 16 + lr;      // 4 waves = 4 N-tiles (N=64)
  v8f acc = {};
  for (int k0 = 0; k0 < 64; k0 += 4) {
    v2f a, bf;
    a.x = sA[lr * 64 + k0 + kh];
    a.y = sA[lr * 64 + k0 + kh + 1];
    bf.x = w0[(size_t)(k0 + kh) * 64 + col];
    bf.y = w0[(size_t)(k0 + kh + 1) * 64 + col];
    acc = wmma_f32_4(a, bf, acc);
  }
  for (int g = 0; g < 8; ++g) {
    int p = mt * 16 + g + rb;
    int b = p >> 8, hw = p & 255;
    attn[(size_t)b * 16384 + (size_t)col * 256 + hw] = acc[g];
  }
}

// residual add, broadcast over l (attention output is l-invariant)
__global__ void k_resadd(float* __restrict__ out, const float* __restrict__ attn) {
  int i = blockIdx.x * 256 + threadIdx.x;
  out[i] += attn[i & 65535];
}

// ===================================================================
// Fused FF over time: X2(65536x32) -> h1(.,128) lrelu -> out += (.,32)
// Both GEMMs via WMMA f32 16x16x4; h1 kept in LDS (no 32MB buffer).
// ===================================================================
__global__ void k_ff(const float* __restrict__ X2,
                     const float* __restrict__ w1, const float* __restrict__ b1,
                     const float* __restrict__ w2, const float* __restrict__ b2,
                     float* __restrict__ out) {
  __shared__ float sX[16 * 32];
  __shared__ float sH[16 * 128];
  __shared__ float sB1[32 * 128];   // B1[l][m] = ff_w1[m][l]
  __shared__ float sB2[128 * 32];   // B2[m][lo] = ff_w2[lo][m]
  int mt = blockIdx.x;              // 0..4095
  int t = threadIdx.x;              // 0..127
  for (int j = 0; j < 4; ++j) sX[t + j * 128] = X2[(size_t)mt * 512 + t + j * 128];
  for (int j = 0; j < 32; ++j) {
    int e = t + j * 128;
    sB1[e] = w1[(size_t)(e & 127) * 32 + (e >> 7)];
  }
  for (int j = 0; j < 32; ++j) {
    int e = t + j * 128;
    sB2[e] = w2[(size_t)(e & 31) * 128 + (e >> 5)];
  }
  __syncthreads();
  int wave = t >> 5, lane = t & 31;
  int lr = lane & 15, kh = (lane >> 4) << 1, rb = (lane >> 4) << 3;
  // phase 1: 16x32 @ 32x128 -> lrelu -> sH
  for (int nt = wave * 2; nt < wave * 2 + 2; ++nt) {
    int col = nt * 16 + lr;
    v8f acc = {};
    for (int k0 = 0; k0 < 32; k0 += 4) {
      v2f a, bf;
      a.x = sX[lr * 32 + k0 + kh];
      a.y = sX[lr * 32 + k0 + kh + 1];
      bf.x = sB1[(k0 + kh) * 128 + col];
      bf.y = sB1[(k0 + kh + 1) * 128 + col];
      acc = wmma_f32_4(a, bf, acc);
    }
    float bias = b1[col];
    for (int g = 0; g < 8; ++g)
      sH[(g + rb) * 128 + col] = lrelu(acc[g] + bias);
  }
  __syncthreads();
  // phase 2: 16x128 @ 128x32 -> out += (residual, scattered over l)
  if (wave < 2) {
    int col = wave * 16 + lr;
    v8f acc = {};
    for (int k0 = 0; k0 < 128; k0 += 4) {
      v2f a, bf;
      a.x = sH[lr * 128 + k0 + kh];
      a.y = sH[lr * 128 + k0 + kh + 1];
      bf.x = sB2[(k0 + kh) * 32 + col];
      bf.y = sB2[(k0 + kh + 1) * 32 + col];
      acc = wmma_f32_4(a, bf, acc);
    }
    float bias = b2[col];
    for (int g = 0; g < 8; ++g) {
      int p = mt * 16 + g + rb;
      out[(size_t)col * 65536 + p] += acc[g] + bias;
    }
  }
}

// ===================================================================
extern "C" void kernel_launch(void* const* d_in, const int* in_sizes, int n_in,
                              void* d_out, int out_size, void* d_ws, size_t ws_size,
                              hipStream_t stream) {
  (void)in_sizes; (void)n_in; (void)out_size; (void)ws_size;
  const float* x       = (const float*)d_in[0];
  const float* gw1     = (const float*)d_in[1];
  const float* gb1     = (const float*)d_in[2];
  const float* gw2     = (const float*)d_in[3];
  const float* gb2     = (const float*)d_in[4];
  const float* gw3     = (const float*)d_in[5];
  const float* gb3     = (const float*)d_in[6];
  const float* gw4     = (const float*)d_in[7];
  const float* gb4     = (const float*)d_in[8];
  const float* conv1_w = (const float*)d_in[9];
  const float* conv1_b = (const float*)d_in[10];
  const float* conv2_w = (const float*)d_in[11];
  // d_in[12] = conv2_b : cancels in softmax (constant over Lk after cq/bias removal)
  // d_in[13] = wq      : q path cancels in softmax entirely
  const float* wk      = (const float*)d_in[14];
  const float* wv      = (const float*)d_in[15];
  const float* w0      = (const float*)d_in[16];
  const float* ln1_w   = (const float*)d_in[17];
  const float* ln1_b   = (const float*)d_in[18];
  const float* ln2_w   = (const float*)d_in[19];
  const float* ln2_b   = (const float*)d_in[20];
  const float* ff_w1   = (const float*)d_in[21];
  const float* ff_b1   = (const float*)d_in[22];
  const float* ff_w2   = (const float*)d_in[23];
  const float* ff_b2   = (const float*)d_in[24];
  float* out = (float*)d_out;
  float* ws  = (float*)d_ws;

  k_weff<<<2 * 128, 64, 0, stream>>>(conv1_w, conv1_b, wk, wv, ws);
  k_featgen<<<128, 256, 0, stream>>>(x, gw1, gb1, gw2, gb2, gw3, gb3, gw4, gb4, out);

  for (int d = 0; d < 2; ++d) {
    k_layernorm<<<128, 256, 0, stream>>>(out, ln1_w + (size_t)d * 16384,
                                         ln1_b + (size_t)d * 16384, ws + OFF_X, 0);
    k_kvgemm<<<2048, 128, 0, stream>>>(ws + OFF_X, ws + OFF_WEFF + (size_t)d * 8192,
                                       ws + OFF_BEFF + (size_t)d * 128,
                                       ws + OFF_K, ws + OFF_V);
    k_headconv<<<512, 256, 0, stream>>>(ws + OFF_K, conv2_w + (size_t)d * 25600,
                                        ws + OFF_CK);
    k_softmax_av<<<8192, 256, 0, stream>>>(ws + OFF_CK, ws + OFF_V, ws + OFF_AP);
    k_w0gemm<<<64, 128, 0, stream>>>(ws + OFF_AP, w0 + (size_t)d * 4096, ws + OFF_AT);
    k_resadd<<<8192, 256, 0, stream>>>(out, ws + OFF_AT);
    k_layernorm<<<128, 256, 0, stream>>>(out, ln2_w + (size_t)d * 16384,
                                         ln2_b + (size_t)d * 16384, ws + OFF_X, 1);
    k_ff<<<4096, 128, 0, stream>>>(ws + OFF_X, ff_w1 + (size_t)d * 4096,
                                   ff_b1 + (size_t)d * 128,
                                   ff_w2 + (size_t)d * 4096,
                                   ff_b2 + (size_t)d * 32, out);
  }
}